// HorizontalConvolution_73048803770964
// MI455X (gfx1250) — compile-verified
//
#include <hip/hip_runtime.h>

// HorizontalConvolution on MI455X (gfx1250, wave32, WMMA + TDM).
//
// Weight-streaming bound: 273 MB fp32 weights, ~12us HBM floor @ 23.3 TB/s.
// Each block's weight slab (256 K x 32 f, row stride F=128) is DMA'd into LDS
// by the Tensor Data Mover (tensor_load_to_lds), double-buffered so the next
// slab streams while the current one feeds V_WMMA_F32_16X16X4_F32.

typedef __attribute__((ext_vector_type(2))) float v2f;
typedef __attribute__((ext_vector_type(8))) float v8f;
typedef __attribute__((ext_vector_type(4))) unsigned int u32x4;
typedef __attribute__((ext_vector_type(8))) unsigned int u32x8;

#define Bsz 64
#define Lsz 64
#define Dsz 256
#define Fsz 128
#define ROWF 40  // LDS floats per K-row: 32 data + 8 pad (TDM pad) =>
                 // rows r, r+2 land on bank sets offset by 80%64=16: the two
                 // lane-halves of each B read are conflict-free.

struct WPtrs { const float* w[Lsz]; };  // one Conv1D kernel per k = 1..64

// TDM: DMA a 256x32-float tile (row stride Fsz floats) from global into LDS,
// padding each 32-DWORD row with 8 DWORDs (LDS row stride = ROWF floats).
__device__ __forceinline__ void tdm_load_slab(const float* gsrc, unsigned lds_byte) {
  const unsigned long long ga = (unsigned long long)(uintptr_t)gsrc;
  // D# group 0: count=1 | lds_addr | global_addr[56:0] | type=2
  u32x4 d0 = {1u, lds_byte, (unsigned)ga, (unsigned)(ga >> 32) | 0x80000000u};
  // D# group 1:
  //  w0: data_size=2(4B)<<16 | pad_en<<20 | pad_interval=4(32DW)<<22 | pad_amount=7(8DW)<<25
  //  w1: tensor_dim0(=32) low16 in bits[63:48]
  //  w2: tensor_dim1(=256) low16 in bits[95:80]
  //  w3: tile_dim0(=32) in bits[127:112]
  //  w4: tile_dim1(=256) in bits[143:128]
  //  w5: tensor_dim0_stride(=128) low32
  u32x8 d1 = {0x0F120000u, 32u << 16, 256u << 16, 32u << 16,
              256u, 128u, 0u, 0u};
  asm volatile("tensor_load_to_lds %0, %1" ::"s"(d0), "s"(d1) : "memory");
}

// One 256-wide K-slab for NT (1 or 2) t positions x 2 f-tiles.
// B from LDS (row-major, ROWF stride), A from global (x is L2-resident).
template <int NT>
__device__ __forceinline__ void slab_accum(v8f (&c)[NT][2],
                                           const float* __restrict__ xA0,
                                           const float* bp, int half, int mn) {
#pragma unroll 2
  for (int kk = 0; kk < Dsz; kk += 4) {
    // B tile rows kk+2*half, kk+2*half+1; cols mn / 16+mn.
    const float* r0 = bp + (kk + 2 * half) * ROWF;
    const v2f b0 = {r0[mn], r0[ROWF + mn]};            // f-tile 0
    const v2f b1 = {r0[16 + mn], r0[ROWF + 16 + mn]};  // f-tile 1
#pragma unroll
    for (int i = 0; i < NT; ++i) {
      // A (16x4 f32): lanes 0-15 K=kk,kk+1; lanes 16-31 K=kk+2,kk+3
      // (xA0 already includes +2*half). t and t+4 are 4*Dsz apart.
      v2f a = *(const v2f*)(xA0 + (size_t)i * 4 * Dsz + kk);
      c[i][0] = __builtin_amdgcn_wmma_f32_16x16x4_f32(false, a, false, b0,
                                                      (short)0, c[i][0], false, false);
      c[i][1] = __builtin_amdgcn_wmma_f32_16x16x4_f32(false, a, false, b1,
                                                      (short)0, c[i][1], false, false);
    }
  }
}

__global__ __launch_bounds__(128)
void hconv_wmma_kernel(const float* __restrict__ X, WPtrs Ws,
                       const float* __restrict__ bs, float* __restrict__ out) {
  const int k  = blockIdx.x + 1;   // n-gram size 1..64
  const int Tk = Lsz - k + 1;      // valid conv positions
  const int t0 = blockIdx.y * 8;   // this block's t-chunk
  if (t0 >= Tk) return;            // uniform exit before any barrier

  const int fB = (blockIdx.z >> 2) * 32;  // 32-wide filter block (2 tiles)
  const int b0 = (blockIdx.z & 3) * 16;   // batch tile

  const int tid  = threadIdx.x;
  const int lane = tid & 31;
  const int wave = tid >> 5;
  const int half = lane >> 4;
  const int mn   = lane & 15;

  // Double-buffered weight slab: 2 x 256 rows x ROWF floats = 80 KB.
  __shared__ __align__(16) float wslab[2][Dsz * ROWF];

  const float* __restrict__ W = Ws.w[k - 1];  // (k, D, F) row-major
  const float* xbase = X + (size_t)(b0 + mn) * Lsz * Dsz + 2 * half;

  const int tw = t0 + wave;                       // this wave's first t
  const int nt = (tw + 4 < Tk) ? 2 : (tw < Tk ? 1 : 0);

  v8f c[2][2] = {};  // [t][f-tile], live across all K-slabs

  // Prologue: DMA slab 0 (wave 0 drives the TDM; EXEC-independent, per-wave).
  if (wave == 0)
    tdm_load_slab(W + fB, (unsigned)(uintptr_t)&wslab[0][0]);

  for (int js = 0; js < k; ++js) {
    if (wave == 0) {
      if (js + 1 < k) {  // kick next slab into the other buffer, then wait
        tdm_load_slab(W + (size_t)(js + 1) * Dsz * Fsz + fB,
                      (unsigned)(uintptr_t)&wslab[(js + 1) & 1][0]);
        __builtin_amdgcn_s_wait_tensorcnt(1);  // slab js complete, js+1 in flight
      } else {
        __builtin_amdgcn_s_wait_tensorcnt(0);
      }
    }
    __syncthreads();  // slab js visible to all waves

    const float* bp  = &wslab[js & 1][0];
    const float* xA0 = xbase + (size_t)(tw + js) * Dsz;
    if (nt == 2)      slab_accum<2>(c, xA0, bp, half, mn);
    else if (nt == 1) { v8f (&c1)[1][2] = (v8f(&)[1][2])c; slab_accum<1>(c1, xA0, bp, half, mn); }
    __syncthreads();  // all reads done before this buffer is re-filled (js+2)
  }

  // Bias + ReLU + max over this wave's t positions.
  const float bias0 = bs[(size_t)(k - 1) * Fsz + fB + mn];
  const float bias1 = bs[(size_t)(k - 1) * Fsz + fB + 16 + mn];
  v8f vmax[2] = {};  // ReLU >= 0: 0 is the max identity (idle waves stay 0)
  for (int i = 0; i < nt; ++i)
#pragma unroll
    for (int r = 0; r < 8; ++r) {
      vmax[0][r] = fmaxf(vmax[0][r], fmaxf(c[i][0][r] + bias0, 0.0f));
      vmax[1][r] = fmaxf(vmax[1][r], fmaxf(c[i][1][r] + bias1, 0.0f));
    }

  // Cross-wave max through LDS (reuse wslab as raw floats).
  float* red = &wslab[0][0];  // [wave][ftile][lane][r] = 2048 floats
#pragma unroll
  for (int ft = 0; ft < 2; ++ft)
#pragma unroll
    for (int r = 0; r < 8; ++r)
      red[(((wave * 2 + ft) * 32) + lane) * 8 + r] = vmax[ft][r];
  __syncthreads();

  if (wave < 2) {  // wave 0 -> f-tile 0, wave 1 -> f-tile 1
    const int ft = wave;
#pragma unroll
    for (int r = 0; r < 8; ++r) {
      float m = red[(((0 * 2 + ft) * 32) + lane) * 8 + r];
      m = fmaxf(m, red[(((1 * 2 + ft) * 32) + lane) * 8 + r]);
      m = fmaxf(m, red[(((2 * 2 + ft) * 32) + lane) * 8 + r]);
      m = fmaxf(m, red[(((3 * 2 + ft) * 32) + lane) * 8 + r]);
      // C layout: VGPR r holds M=r (lanes 0-15) / M=r+8 (lanes 16-31), N=mn.
      const int brow = b0 + r + 8 * half;
      out[(size_t)brow * (Lsz * Fsz) + (size_t)(k - 1) * Fsz + fB + ft * 16 + mn] = m;
    }
  }
}

extern "C" void kernel_launch(void* const* d_in, const int* in_sizes, int n_in,
                              void* d_out, int out_size, void* d_ws, size_t ws_size,
                              hipStream_t stream) {
  (void)in_sizes; (void)d_ws; (void)ws_size; (void)out_size;

  const float* X = (const float*)d_in[0];
  WPtrs Ws;
  for (int i = 0; i < Lsz; ++i) Ws.w[i] = (const float*)d_in[1 + i];
  const float* bs = (const float*)d_in[n_in - 1];
  float* out = (float*)d_out;

  dim3 grid(Lsz /*k*/, 8 /*t-chunks*/, 16 /*4 f-blocks x 4 b-tiles*/);
  dim3 block(128);  // 4 wave32
  hconv_wmma_kernel<<<grid, block, 0, stream>>>(X, Ws, bs, out);
}